// HyperGNN_79130477462184
// MI455X (gfx1250) — compile-verified
//
#include <hip/hip_runtime.h>
#include <math.h>

// ---------------------------------------------------------------------------
// HGCN (Poincare ball, c=1) full pipeline for MI455X / gfx1250.
// fp32 throughout (matches reference numerics); GEMMs via V_WMMA_F32_16X16X4_F32.
// Edge scatter uses native fp32 global atomics (L2-resident: node tensor = 12.8MB).
// ---------------------------------------------------------------------------

constexpr int NN   = 50000;    // nodes
constexpr int NE   = 1200000;  // edges
constexpr int NG   = 128;      // graphs
constexpr int DIN  = 32;
constexpr int DH   = 64;
constexpr int DOUT = 10;

#define EPSC  1e-7f
#define MAXN  (1.0f - 1e-5f)

typedef float v2f __attribute__((ext_vector_type(2)));
typedef float v8f __attribute__((ext_vector_type(8)));

__device__ __forceinline__ void fatomic_add(float* p, float v) {
    // native global_atomic_add_f32 (no CAS loop)
    unsafeAtomicAdd(p, v);
}

__device__ __forceinline__ float wave_sum32(float v) {
#pragma unroll
    for (int o = 16; o > 0; o >>= 1) v += __shfl_xor(v, o, 32);
    return v;
}

// combined scale for logmap0(proj(expmap0(h))) given the true norm tn = ||h||
__device__ __forceinline__ float chain_lpe(float tn) {
    float n1 = fmaxf(tn, EPSC);
    float s1 = tanhf(n1) / n1;          // expmap0
    float ne = tn * s1;                 // norm after expmap0
    float n2 = fmaxf(ne, EPSC);
    float s2 = (n2 > MAXN) ? (MAXN / n2) : 1.0f;   // proj
    float np = ne * s2;
    float n3 = fminf(fmaxf(np, EPSC), MAXN);
    float s3 = atanhf(n3) / n3;         // logmap0
    return s1 * s2 * s3;
}

// combined scale for proj(expmap0(h))
__device__ __forceinline__ float chain_pe(float tn) {
    float n1 = fmaxf(tn, EPSC);
    float s1 = tanhf(n1) / n1;
    float ne = tn * s1;
    float n2 = fmaxf(ne, EPSC);
    float s2 = (n2 > MAXN) ? (MAXN / n2) : 1.0f;
    return s1 * s2;
}

// ---------------------------------------------------------------------------
__global__ void k_zero(float* __restrict__ p, int n) {
    for (int i = blockIdx.x * blockDim.x + threadIdx.x; i < n;
         i += gridDim.x * blockDim.x)
        p[i] = 0.0f;
}

__global__ void k_counts(const int* __restrict__ dst, float* __restrict__ cnt, int e_cnt) {
    int e = blockIdx.x * blockDim.x + threadIdx.x;
    if (e >= e_cnt) return;
    fatomic_add(&cnt[dst[e]], 1.0f);
}

__global__ void k_gcnt(const int* __restrict__ batch, float* __restrict__ gcnt, int n) {
    int i = blockIdx.x * blockDim.x + threadIdx.x;
    if (i >= n) return;
    fatomic_add(&gcnt[batch[i]], 1.0f);
}

// t0 = logmap0(proj(expmap0(x))), x: [NN, 32]; one wave per node, lane = feature
__global__ void k_prelude(const float* __restrict__ X, float* __restrict__ T, int n) {
    int tid  = blockIdx.x * blockDim.x + threadIdx.x;
    int node = tid >> 5;
    if (node >= n) return;
    int lane = threadIdx.x & 31;
    float v  = X[node * DIN + lane];
    float tn = sqrtf(wave_sum32(v * v));
    T[node * DIN + lane] = v * chain_lpe(tn);
}

// H[N,64] = T[N,K] @ W[K,64] + b.  One wave -> 16 rows x 64 cols, K-loop of 4.
// A-frag (f32 16x4): lanes 0-15 hold K=0,1 ; lanes 16-31 hold K=2,3 (row = lane&15)
// B-frag (f32 4x16): VGPR0/1 hold K rows {0,1} (lanes 0-15) / {2,3} (lanes 16-31)
// C: VGPR r = row mbase+r (lanes 0-15) / mbase+r+8 (lanes 16-31), col = lane&15
template <int K>
__global__ __launch_bounds__(256) void k_gemm(const float* __restrict__ T,
                                              const float* __restrict__ W,
                                              const float* __restrict__ bias,
                                              float* __restrict__ H, int n) {
    __shared__ float Ws[K * 64];
    __shared__ float Bs[64];
    int t = threadIdx.x;
    for (int i = t; i < K * 64; i += 256) Ws[i] = W[i];
    if (t < 64) Bs[t] = bias[t];
    __syncthreads();

    int wave  = t >> 5;
    int lane  = t & 31;
    int mbase = blockIdx.x * 128 + wave * 16;
    if (mbase >= n) return;                    // NN % 16 == 0: no partial tiles

    int nl   = lane & 15;
    int koff = (lane >> 4) * 2;
    const float* arow = T + (size_t)(mbase + nl) * K + koff;

    v8f acc0 = {0,0,0,0,0,0,0,0};
    v8f acc1 = {0,0,0,0,0,0,0,0};
    v8f acc2 = {0,0,0,0,0,0,0,0};
    v8f acc3 = {0,0,0,0,0,0,0,0};

#pragma unroll
    for (int k = 0; k < K; k += 4) {
        v2f a;
        a.x = arow[k];
        a.y = arow[k + 1];
        const float* b0 = Ws + (k + koff) * 64;
        const float* b1 = b0 + 64;
        v2f bb;
        bb.x = b0[nl +  0]; bb.y = b1[nl +  0];
        acc0 = __builtin_amdgcn_wmma_f32_16x16x4_f32(false, a, false, bb, (short)0, acc0, false, false);
        bb.x = b0[nl + 16]; bb.y = b1[nl + 16];
        acc1 = __builtin_amdgcn_wmma_f32_16x16x4_f32(false, a, false, bb, (short)0, acc1, false, false);
        bb.x = b0[nl + 32]; bb.y = b1[nl + 32];
        acc2 = __builtin_amdgcn_wmma_f32_16x16x4_f32(false, a, false, bb, (short)0, acc2, false, false);
        bb.x = b0[nl + 48]; bb.y = b1[nl + 48];
        acc3 = __builtin_amdgcn_wmma_f32_16x16x4_f32(false, a, false, bb, (short)0, acc3, false, false);
    }

    int mo = (lane >> 4) * 8;
#pragma unroll
    for (int r = 0; r < 8; ++r) {
        float* o = H + (size_t)(mbase + mo + r) * 64 + nl;
        o[0]  = acc0[r] + Bs[nl];
        o[16] = acc1[r] + Bs[nl + 16];
        o[32] = acc2[r] + Bs[nl + 32];
        o[48] = acc3[r] + Bs[nl + 48];
    }
}

// scatter-add: ACC[dst[e]] += H[src[e]]; one wave per edge, float2 per lane
__global__ void k_aggregate(const int* __restrict__ src, const int* __restrict__ dst,
                            const float* __restrict__ H, float* __restrict__ ACC,
                            int e_cnt) {
    int tid = blockIdx.x * blockDim.x + threadIdx.x;
    int e   = tid >> 5;
    if (e >= e_cnt) return;
    int lane = tid & 31;
    if (lane == 0 && e + 2048 < e_cnt) {        // global_prefetch_b8 on edge lists
        __builtin_prefetch(src + e + 2048, 0, 1);
        __builtin_prefetch(dst + e + 2048, 0, 1);
    }
    int f = lane * 2;
    int s = src[e];
    int d = dst[e];
    float2 v = *(const float2*)(H + (size_t)s * 64 + f);
    float* a = ACC + (size_t)d * 64 + f;
    fatomic_add(a,     v.x);
    fatomic_add(a + 1, v.y);
}

// mean -> (leaky_relu) -> expmap0 -> proj -> logmap0 ; wave per node (in-place ok)
__global__ void k_post(const float* __restrict__ ACC, const float* __restrict__ CNT,
                       float* __restrict__ T, int n, int applyAct) {
    int tid  = blockIdx.x * blockDim.x + threadIdx.x;
    int node = tid >> 5;
    if (node >= n) return;
    int lane = threadIdx.x & 31;
    int f    = lane * 2;
    float c  = fmaxf(CNT[node], 1.0f);
    float2 v = *(const float2*)(ACC + (size_t)node * 64 + f);
    v.x /= c;
    v.y /= c;
    if (applyAct) {
        v.x = (v.x >= 0.0f) ? v.x : 0.2f * v.x;
        v.y = (v.y >= 0.0f) ? v.y : 0.2f * v.y;
    }
    float tn = sqrtf(wave_sum32(v.x * v.x + v.y * v.y));
    float s  = chain_lpe(tn);
    float2 o;
    o.x = v.x * s;
    o.y = v.y * s;
    *(float2*)(T + (size_t)node * 64 + f) = o;
}

// POOL[batch[i]] += T[i]; wave per node
__global__ void k_pool(const float* __restrict__ T, const int* __restrict__ batch,
                       float* __restrict__ POOL, int n) {
    int tid  = blockIdx.x * blockDim.x + threadIdx.x;
    int node = tid >> 5;
    if (node >= n) return;
    int lane = tid & 31;
    int f    = lane * 2;
    int g    = batch[node];
    float2 v = *(const float2*)(T + (size_t)node * 64 + f);
    float* p = POOL + (size_t)g * 64 + f;
    fatomic_add(p,     v.x);
    fatomic_add(p + 1, v.y);
}

// per-graph head: mean -> proj(expmap0) -> logmap0 -> @Wl + bl -> proj(expmap0)
__global__ void k_final(const float* __restrict__ POOL, const float* __restrict__ GCNT,
                        const float* __restrict__ Wl, const float* __restrict__ bl,
                        float* __restrict__ out) {
    int g = blockIdx.x * blockDim.x + threadIdx.x;
    if (g >= NG) return;
    float c = fmaxf(GCNT[g], 1.0f);
    float u[DH];
    float ss = 0.0f;
#pragma unroll
    for (int i = 0; i < DH; ++i) {
        u[i] = POOL[(size_t)g * DH + i] / c;
        ss += u[i] * u[i];
    }
    float s = chain_lpe(sqrtf(ss));        // logmap0(proj(expmap0(pooled)))
#pragma unroll
    for (int i = 0; i < DH; ++i) u[i] *= s;

    float v[DOUT];
#pragma unroll
    for (int j = 0; j < DOUT; ++j) v[j] = bl[j];
    for (int i = 0; i < DH; ++i) {
        float ui = u[i];
#pragma unroll
        for (int j = 0; j < DOUT; ++j) v[j] += ui * Wl[i * DOUT + j];
    }
    float ss2 = 0.0f;
#pragma unroll
    for (int j = 0; j < DOUT; ++j) ss2 += v[j] * v[j];
    float s2 = chain_pe(sqrtf(ss2));       // proj(expmap0(z))
#pragma unroll
    for (int j = 0; j < DOUT; ++j) out[g * DOUT + j] = v[j] * s2;
}

// ---------------------------------------------------------------------------
extern "C" void kernel_launch(void* const* d_in, const int* in_sizes, int n_in,
                              void* d_out, int out_size, void* d_ws, size_t ws_size,
                              hipStream_t stream) {
    const float* X     = (const float*)d_in[0];
    const int*   EI    = (const int*)d_in[1];   // int per harness convention
    const int*   SRC   = EI;
    const int*   DST   = EI + NE;
    const int*   BATCH = (const int*)d_in[2];
    const float* W1 = (const float*)d_in[3];
    const float* B1 = (const float*)d_in[4];
    const float* W2 = (const float*)d_in[5];
    const float* B2 = (const float*)d_in[6];
    const float* W3 = (const float*)d_in[7];
    const float* B3 = (const float*)d_in[8];
    const float* WL = (const float*)d_in[9];
    const float* BL = (const float*)d_in[10];
    float* OUT = (float*)d_out;

    // workspace layout (fp32):
    //   T    [NN*64]  tangent node features; also aliased as aggregation ACC
    //   H    [NN*64]  GEMM output (gather source)
    //   CNT  [NN]     in-degree counts
    //   POOL [NG*64]  pooled sums
    //   GCNT [NG]     nodes per graph
    float* T    = (float*)d_ws;
    float* H    = T + (size_t)NN * DH;
    float* CNT  = H + (size_t)NN * DH;
    float* POOL = CNT + NN;
    float* GCNT = POOL + (size_t)NG * DH;
    float* ACC  = T;   // safe: GEMM consumes T before ACC is zeroed (stream order)

    const int nodeWaveBlk = (NN * 32 + 255) / 256;
    const int edgeWaveBlk = (int)(((long long)NE * 32 + 255) / 256);
    const int gemmBlk     = (NN + 127) / 128;

    // counts (recomputed every call; deterministic)
    k_zero<<<256, 256, 0, stream>>>(CNT, NN + NG * DH + NG);
    k_counts<<<(NE + 255) / 256, 256, 0, stream>>>(DST, CNT, NE);
    k_gcnt<<<(NN + 255) / 256, 256, 0, stream>>>(BATCH, GCNT, NN);

    // t0 = logmap0(proj(expmap0(x)))
    k_prelude<<<nodeWaveBlk, 256, 0, stream>>>(X, T, NN);

    // ---- layer 1 (K=32, leaky_relu) ----
    k_gemm<DIN><<<gemmBlk, 256, 0, stream>>>(T, W1, B1, H, NN);
    k_zero<<<512, 256, 0, stream>>>(ACC, NN * DH);
    k_aggregate<<<edgeWaveBlk, 256, 0, stream>>>(SRC, DST, H, ACC, NE);
    k_post<<<nodeWaveBlk, 256, 0, stream>>>(ACC, CNT, T, NN, 1);

    // ---- layer 2 (K=64, leaky_relu) ----
    k_gemm<DH><<<gemmBlk, 256, 0, stream>>>(T, W2, B2, H, NN);
    k_zero<<<512, 256, 0, stream>>>(ACC, NN * DH);
    k_aggregate<<<edgeWaveBlk, 256, 0, stream>>>(SRC, DST, H, ACC, NE);
    k_post<<<nodeWaveBlk, 256, 0, stream>>>(ACC, CNT, T, NN, 1);

    // ---- layer 3 (K=64, no act) ----
    k_gemm<DH><<<gemmBlk, 256, 0, stream>>>(T, W3, B3, H, NN);
    k_zero<<<512, 256, 0, stream>>>(ACC, NN * DH);
    k_aggregate<<<edgeWaveBlk, 256, 0, stream>>>(SRC, DST, H, ACC, NE);
    k_post<<<nodeWaveBlk, 256, 0, stream>>>(ACC, CNT, T, NN, 0);

    // ---- pooling + head ----
    k_pool<<<nodeWaveBlk, 256, 0, stream>>>(T, BATCH, POOL, NN);
    k_final<<<1, 128, 0, stream>>>(POOL, GCNT, WL, BL, OUT);
}